// Ensemble_47665547051123
// MI455X (gfx1250) — compile-verified
//
#include <hip/hip_runtime.h>
#include <stdint.h>

// Problem geometry
#define S          16384        // 128*128 neurons
#define NTILES     8            // column tiles, 2048 cols each
#define COLS_BLK   2048
#define NKCHUNK    64           // K chunks -> partial-sum rows in d_ws
#define KC         256          // rows per K chunk (64*256 = 16384)

typedef __attribute__((ext_vector_type(2))) float v2f;
typedef __attribute__((ext_vector_type(4))) float v4f;
typedef __attribute__((ext_vector_type(8))) float v8f;

// -----------------------------------------------------------------------------
// Kernel 1: bandwidth-bound sparse row-sum of lateral_weights.
//   P[kc][n] = sum_{k in chunk kc, spk[k]!=0} W[k][n]
//
// Phase A: ballot/prefix compaction of active row indices into LDS
//          (deterministic, ascending-k order -> fixed FP summation order).
// Phase B: branch-free unroll-4 stream over active rows; row base addresses
//          forced wave-uniform with readfirstlane so loads take the
//          saddr+voffset form and can be clause-pipelined; prefetch ahead.
// -----------------------------------------------------------------------------
__global__ __launch_bounds__(256) void lateral_partial_kernel(
    const float* __restrict__ W,
    const unsigned char* __restrict__ spk,   // jnp.bool_ -> 1 byte/elem
    float* __restrict__ P)
{
    __shared__ int s_idx[KC];                // compacted active row indices
    __shared__ int s_cnt[8];                 // per-wave active counts

    const int tile = blockIdx.x;             // 0..7
    const int kc   = blockIdx.y;             // 0..63
    const int t    = threadIdx.x;            // 0..255
    const int wv   = t >> 5;
    const int lane = t & 31;
    const int k0   = kc * KC;

    // ---- Phase A: compaction (one row per thread) ----
    const bool pred = (spk[k0 + t] != 0);
    const uint32_t mask = __builtin_amdgcn_ballot_w32(pred);
    const int pfx = __builtin_popcount(mask & ((1u << lane) - 1u));
    if (lane == 0) s_cnt[wv] = __builtin_popcount(mask);
    __syncthreads();

    int off = 0;
#pragma unroll
    for (int w = 0; w < 8; ++w)
        if (w < wv) off += s_cnt[w];
    int nact = 0;
#pragma unroll
    for (int w = 0; w < 8; ++w) nact += s_cnt[w];

    if (pred) s_idx[off + pfx] = k0 + t;
    __syncthreads();

    // ---- Phase B: dense stream over active rows ----
    const int col0 = tile * COLS_BLK + t * 4;
    const int col1 = col0 + 1024;

    v4f a0 = {0.f, 0.f, 0.f, 0.f};
    v4f a1 = {0.f, 0.f, 0.f, 0.f};

    int i = 0;
    for (; i + 4 <= nact; i += 4) {
        const int ka = __builtin_amdgcn_readfirstlane(s_idx[i + 0]);
        const int kb = __builtin_amdgcn_readfirstlane(s_idx[i + 1]);
        const int kx = __builtin_amdgcn_readfirstlane(s_idx[i + 2]);
        const int kd = __builtin_amdgcn_readfirstlane(s_idx[i + 3]);
        const float* ra = W + (size_t)ka * S;
        const float* rb = W + (size_t)kb * S;
        const float* rc = W + (size_t)kx * S;
        const float* rd = W + (size_t)kd * S;

        // 8 independent b128 loads in flight before any accumulate
        const v4f wa0 = *(const v4f*)(ra + col0);
        const v4f wa1 = *(const v4f*)(ra + col1);
        const v4f wb0 = *(const v4f*)(rb + col0);
        const v4f wb1 = *(const v4f*)(rb + col1);
        const v4f wc0 = *(const v4f*)(rc + col0);
        const v4f wc1 = *(const v4f*)(rc + col1);
        const v4f wd0 = *(const v4f*)(rd + col0);
        const v4f wd1 = *(const v4f*)(rd + col1);

        if (i + 16 < nact) {                 // gfx1250 global_prefetch_b8
            const int kp = __builtin_amdgcn_readfirstlane(s_idx[i + 16]);
            __builtin_prefetch(W + (size_t)kp * S + col0, 0, 0);
            __builtin_prefetch(W + (size_t)kp * S + col1, 0, 0);
        }

        // ascending-k accumulation order (deterministic)
        a0 += wa0; a1 += wa1;
        a0 += wb0; a1 += wb1;
        a0 += wc0; a1 += wc1;
        a0 += wd0; a1 += wd1;
    }
    for (; i < nact; ++i) {
        const int ka = __builtin_amdgcn_readfirstlane(s_idx[i]);
        const float* ra = W + (size_t)ka * S;
        a0 += *(const v4f*)(ra + col0);
        a1 += *(const v4f*)(ra + col1);
    }

    float* o = P + (size_t)kc * S;
    *(v4f*)(o + col0) = a0;
    *(v4f*)(o + col1) = a1;
}

// -----------------------------------------------------------------------------
// Kernel 2: reduce the 64 partial rows per column with V_WMMA_F32_16X16X4_F32
// (A = all-ones 16x4, B = 4x16 slab of partials -> every D row is the column
// sums; fp32 WMMA so bit-compatible with scalar fp32 adds), then the fused
// neuron update epilogue.
//
// One wave per 16-column tile. B-matrix layout (4x16 f32, 2 VGPRs, wave32):
//   VGPR0: lanes 0-15 = (K=0, N=lane), lanes 16-31 = (K=2, N=lane-16)
//   VGPR1: lanes 0-15 = (K=1, N=lane), lanes 16-31 = (K=3, N=lane-16)
// D layout: VGPR0, lanes 0-15 = (M=0, N=lane)  -> that's our result.
// -----------------------------------------------------------------------------
__global__ __launch_bounds__(256) void reduce_update_kernel(
    const float* __restrict__ P,
    const float* __restrict__ x,
    const float* __restrict__ act,
    const float* __restrict__ thr,
    const float* __restrict__ freq,
    float* __restrict__ out)
{
    const int gid   = blockIdx.x * blockDim.x + threadIdx.x;
    const int wave  = gid >> 5;              // 0..1023, one 16-col tile each
    const int lane  = threadIdx.x & 31;
    const int n0    = wave * 16;
    const int laneN = lane & 15;
    const int half  = lane >> 4;             // 0: K=0/1, 1: K=2/3

    const v2f A = {1.0f, 1.0f};              // ones => D rows = column sums
    v8f acc = {};

#pragma unroll
    for (int s = 0; s < NKCHUNK / 4; ++s) {
        const int c  = s * 4;
        const int kx = c + (half ? 2 : 0);
        const int ky = c + (half ? 3 : 1);
        v2f B;
        B.x = P[(size_t)kx * S + n0 + laneN];
        B.y = P[(size_t)ky * S + n0 + laneN];
        acc = __builtin_amdgcn_wmma_f32_16x16x4_f32(
            /*neg_a=*/false, A, /*neg_b=*/false, B,
            /*c_mod=*/(short)0, acc, /*reuse_a=*/false, /*reuse_b=*/false);
    }

    if (lane < 16) {
        const int n    = n0 + laneN;
        const float lat = acc[0];            // (M=0, N=laneN)
        const float xt  = x[n] + lat;
        float na        = 0.9f * act[n] + xt;        // BETA * a + x_tot
        const float th  = thr[n];
        const bool ns   = na > th;
        const float nf  = 0.95f * freq[n] + 0.05f * (ns ? 1.0f : 0.0f);
        const float nth = (nf > 0.2f) ? (th + 0.05f)
                        : ((nf < 0.2f) ? (th / 1.05f) : th);
        if (ns) na = 0.0f;                   // zero-reset at spiking neurons
        out[0 * S + n] = ns ? 1.0f : 0.0f;   // new_spikes
        out[1 * S + n] = na;                 // new_activation
        out[2 * S + n] = nth;                // new_threshold
        out[3 * S + n] = nf;                 // new_freq
    }
}

// -----------------------------------------------------------------------------
extern "C" void kernel_launch(void* const* d_in, const int* in_sizes, int n_in,
                              void* d_out, int out_size, void* d_ws, size_t ws_size,
                              hipStream_t stream)
{
    const float*         x    = (const float*)d_in[0];
    const float*         act  = (const float*)d_in[1];
    const float*         thr  = (const float*)d_in[2];
    const float*         freq = (const float*)d_in[3];
    const float*         W    = (const float*)d_in[4];
    const unsigned char* spk  = (const unsigned char*)d_in[5];
    float*               out  = (float*)d_out;
    float*               P    = (float*)d_ws;   // needs 64*16384*4 = 4 MiB

    dim3 gA(NTILES, NKCHUNK);                   // 512 blocks
    lateral_partial_kernel<<<gA, 256, 0, stream>>>(W, spk, P);

    // 1024 waves = 1024 column tiles of 16
    reduce_update_kernel<<<128, 256, 0, stream>>>(P, x, act, thr, freq, out);
}